// UncertaintyAwareFusion_41781441855930
// MI455X (gfx1250) — compile-verified
//
#include <hip/hip_runtime.h>
#include <stdint.h>

// UncertaintyAwareFusion: softmax over V=4 views + weighted fuse + sigmoid.
// Pure HBM-bound streaming kernel (624 MB @ 23.3 TB/s ~= 27 us floor).
// Data path: CDNA5 async global->LDS b128 loads, double buffered, per-wave
// s_wait_asynccnt pipelining; NT b128 stores for the streamed outputs.
// Math path: v_exp_f32 + v_rcp_f32 (TRANS) to keep VALU far below the
// memory floor (IEEE div sequences were ~80 VALU per group).

typedef float v4f __attribute__((ext_vector_type(4)));

#define NVIEW 4
#define PTOT  12000000      // B*N = 500000*24
#define PG    3000000       // PTOT/4 float4 groups per view
#define TPB   256
#define FULL_TILES (PG / TPB)            // 11718
#define TAIL_START (FULL_TILES * TPB)    // 2999808
#define TAIL_COUNT (PG - TAIL_START)     // 192
#define SLOT_BYTES (TPB * 16)            // one view-slot per buffer
#define BUF_V4     (8 * TPB)             // 8 slots (4 z + 4 rho) of float4
#define BUF_BYTES  (BUF_V4 * 16)         // 32 KB per buffer, 64 KB double

__device__ __forceinline__ void async_ld_b128(unsigned lds_byte, unsigned voff_byte,
                                              unsigned long long base) {
  // CDNA5 GLOBAL_LOAD_ASYNC_TO_LDS_B128, GVS mode: saddr(64b base) + per-lane
  // 32-bit vaddr offset; first operand is the per-lane LDS byte address.
  asm volatile("global_load_async_to_lds_b128 %0, %1, %2"
               :: "v"(lds_byte), "v"(voff_byte), "s"(base)
               : "memory");
}

__device__ __forceinline__ void wait_async_le8(bool keep8) {
  if (keep8) asm volatile("s_wait_asynccnt 8" ::: "memory");
  else       asm volatile("s_wait_asynccnt 0" ::: "memory");
}

__device__ __forceinline__ void fuse_point(const v4f zv[NVIEW], const v4f rv[NVIEW],
                                           v4f& c4, v4f vv[NVIEW]) {
#pragma unroll
  for (int j = 0; j < 4; ++j) {
    float r0 = rv[0][j], r1 = rv[1][j], r2 = rv[2][j], r3 = rv[3][j];
    float m  = fmaxf(fmaxf(r0, r1), fmaxf(r2, r3));
    float e0 = __expf(r0 - m), e1 = __expf(r1 - m);
    float e2 = __expf(r2 - m), e3 = __expf(r3 - m);
    float inv = __builtin_amdgcn_rcpf(e0 + e1 + e2 + e3);   // v_rcp_f32
    float w0 = e0 * inv, w1 = e1 * inv, w2 = e2 * inv, w3 = e3 * inv;
    float logit = w0 * zv[0][j] + w1 * zv[1][j] + w2 * zv[2][j] + w3 * zv[3][j];
    c4[j] = __builtin_amdgcn_rcpf(1.0f + __expf(-logit));   // sigmoid via v_rcp_f32
    vv[0][j] = w0; vv[1][j] = w1; vv[2][j] = w2; vv[3][j] = w3;
  }
}

__device__ __forceinline__ void store_point(float* __restrict__ out, unsigned g,
                                            const v4f& c4, const v4f vv[NVIEW]) {
  v4f* oc = (v4f*)out;
  __builtin_nontemporal_store(c4, oc + g);
#pragma unroll
  for (int v = 0; v < NVIEW; ++v)
    __builtin_nontemporal_store(vv[v], oc + (unsigned)(1 + v) * PG + g);
}

__global__ void __launch_bounds__(TPB)
fusion_async_kernel(const float* __restrict__ z, const float* __restrict__ rho,
                    float* __restrict__ out) {
  extern __shared__ v4f smem[];   // 2 * 32KB double buffer, dyn-LDS base = 0
  const unsigned t = threadIdx.x;
  const unsigned long long zb = (unsigned long long)(uintptr_t)z;
  const unsigned long long rb = (unsigned long long)(uintptr_t)rho;

  // ---- full tiles: async double-buffered pipeline (uniform per block) ----
  unsigned tile  = blockIdx.x;
  bool     valid = tile < FULL_TILES;
  unsigned buf   = 0;

  auto issue = [&](unsigned tl, unsigned bf) {
    const unsigned g16  = (tl * TPB + t) * 16u;        // byte offset in a view
    const unsigned ldsb = bf * BUF_BYTES + t * 16u;
#pragma unroll
    for (int v = 0; v < NVIEW; ++v) {
      const unsigned voff = (unsigned)v * (unsigned)(PTOT * 4) + g16; // 48MB view stride
      async_ld_b128(ldsb + (unsigned)v * SLOT_BYTES,           voff, zb);
      async_ld_b128(ldsb + (unsigned)(NVIEW + v) * SLOT_BYTES, voff, rb);
    }
  };

  if (valid) issue(tile, buf);

  while (valid) {
    const unsigned ntile  = tile + gridDim.x;
    const bool     nvalid = ntile < FULL_TILES;
    if (nvalid) issue(ntile, buf ^ 1u);     // prefetch next tile into other buffer
    wait_async_le8(nvalid);                 // current buffer's 8 loads retired

    const unsigned bb = buf * BUF_V4 + t;
    v4f zv[NVIEW], rv[NVIEW];
#pragma unroll
    for (int v = 0; v < NVIEW; ++v) {
      zv[v] = smem[bb + (unsigned)v * TPB];
      rv[v] = smem[bb + (unsigned)(NVIEW + v) * TPB];
    }

    v4f c4, vv[NVIEW];
    fuse_point(zv, rv, c4, vv);
    store_point(out, tile * TPB + t, c4, vv);

    tile = ntile; valid = nvalid; buf ^= 1u;
  }

  // ---- tail groups (192): plain guarded b128 path, no async/EXEC issues ----
  const unsigned gtid = blockIdx.x * TPB + t;
  if (gtid < TAIL_COUNT) {
    const unsigned g  = TAIL_START + gtid;
    const v4f* zp = (const v4f*)z;
    const v4f* rp = (const v4f*)rho;
    v4f zv[NVIEW], rv[NVIEW];
#pragma unroll
    for (int v = 0; v < NVIEW; ++v) {
      zv[v] = zp[(unsigned)v * PG + g];
      rv[v] = rp[(unsigned)v * PG + g];
    }
    v4f c4, vv[NVIEW];
    fuse_point(zv, rv, c4, vv);
    store_point(out, g, c4, vv);
  }
}

extern "C" void kernel_launch(void* const* d_in, const int* in_sizes, int n_in,
                              void* d_out, int out_size, void* d_ws, size_t ws_size,
                              hipStream_t stream) {
  (void)in_sizes; (void)n_in; (void)d_ws; (void)ws_size; (void)out_size;
  const float* z   = (const float*)d_in[0];
  const float* rho = (const float*)d_in[1];
  float*       out = (float*)d_out;

  dim3 block(TPB);
  dim3 grid(1024);    // ~11.4 tiles/block -> double-buffer pipeline amortized
  fusion_async_kernel<<<grid, block, 2 * BUF_BYTES, stream>>>(z, rho, out);
}